// AVWDCRNN_3616362463683
// MI455X (gfx1250) — compile-verified
//
#include <hip/hip_runtime.h>
#include <math.h>

// AGCRN (AVWDCRNN) on MI455X / gfx1250, fp32 end-to-end, WMMA f32 16x16x4.
// Pipeline per layer/timestep:
//   build_xs -> gemm (A@xs) -> apply gate (sigmoid) -> build_cnd ->
//   gemm (A@cnd) -> apply cand (tanh) -> update(h,out)
// Node-specific weights are never materialized: apply kernel accumulates
// sum_e E[n,e] * (x_g @ Wpool[e]) with the scalar folded into the WMMA
// A-fragment, so the weight GEMM runs from L2-resident pool weights.
// The diffusion GEMM double-buffers its LDS tiles with CDNA5 async
// global->LDS copies (GLOBAL_LOAD_ASYNC_TO_LDS_B128 / s_wait_asynccnt).

#define N_NODE 2048
#define EMB    16
#define BATCH  16
#define TT     24
#define HID    64
#define CHEBK  2

typedef __attribute__((ext_vector_type(2))) float v2f;
typedef __attribute__((ext_vector_type(8))) float v8f;

// CDNA5 async global->LDS copy, 16 bytes per lane, tracked by ASYNCcnt.
__device__ __forceinline__
void async_copy_b128(unsigned lds_off, unsigned gof, const float* base) {
    asm volatile("global_load_async_to_lds_b128 %0, %1, %2"
                 :: "v"(lds_off), "v"(gof), "s"(base)
                 : "memory");
}

// ---------------------------------------------------------------- supports
// A = softmax(relu(E E^T)) row-wise. 2048x2048, tiny FLOPs -> VALU kernel.
__global__ void __launch_bounds__(256)
supports_kernel(const float* __restrict__ E, float* __restrict__ A) {
    const int n   = blockIdx.x;
    const int tid = threadIdx.x;
    __shared__ float red[256];
    __shared__ float erow[EMB];
    if (tid < EMB) erow[tid] = E[n * EMB + tid];
    __syncthreads();

    float vals[8];
    float lmax = -1e30f;
    for (int j = 0; j < 8; ++j) {
        const int m = tid + j * 256;
        float acc = 0.f;
#pragma unroll
        for (int e = 0; e < EMB; ++e) acc += erow[e] * E[m * EMB + e];
        acc = acc > 0.f ? acc : 0.f;
        vals[j] = acc;
        lmax = fmaxf(lmax, acc);
    }
    red[tid] = lmax; __syncthreads();
    for (int s = 128; s > 0; s >>= 1) {
        if (tid < s) red[tid] = fmaxf(red[tid], red[tid + s]);
        __syncthreads();
    }
    const float rmax = red[0]; __syncthreads();

    float lsum = 0.f;
    for (int j = 0; j < 8; ++j) { vals[j] = expf(vals[j] - rmax); lsum += vals[j]; }
    red[tid] = lsum; __syncthreads();
    for (int s = 128; s > 0; s >>= 1) {
        if (tid < s) red[tid] += red[tid + s];
        __syncthreads();
    }
    const float inv = 1.0f / red[0];
    for (int j = 0; j < 8; ++j) A[n * N_NODE + tid + j * 256] = vals[j] * inv;
}

// ---------------------------------------------------------------- h init
__global__ void init_h_kernel(const float* __restrict__ init_state,
                              float* __restrict__ h, int layer) {
    const int idx = blockIdx.x * blockDim.x + threadIdx.x;
    if (idx >= N_NODE * BATCH * HID) return;
    const int o = idx % HID;
    const int b = (idx / HID) % BATCH;
    const int n = idx / (HID * BATCH);
    // internal h layout: [n][b][o]
    h[idx] = init_state[((layer * BATCH + b) * N_NODE + n) * HID + o];
}

// ---------------------------------------------------------------- xs build
// xs[n][col], col=b*C+c, padded to BCpad with zeros. xs = concat(x_t, h).
__global__ void build_xs_kernel(const float* __restrict__ xsrc, int Cin, int t,
                                const float* __restrict__ h,
                                float* __restrict__ xs, int C, int BCpad) {
    const int idx = blockIdx.x * blockDim.x + threadIdx.x;
    if (idx >= N_NODE * BCpad) return;
    const int n = idx / BCpad;
    const int col = idx % BCpad;
    float v = 0.f;
    if (col < BATCH * C) {
        const int b = col / C, c = col % C;
        if (c < Cin) v = xsrc[((b * TT + t) * N_NODE + n) * Cin + c];
        else         v = h[(n * BATCH + b) * HID + (c - Cin)];
    }
    xs[idx] = v;
}

// cnd = concat(x_t, z*h), z = zr[...,:HID]
__global__ void build_cnd_kernel(const float* __restrict__ xsrc, int Cin, int t,
                                 const float* __restrict__ h,
                                 const float* __restrict__ zr,
                                 float* __restrict__ cnd, int C, int BCpad) {
    const int idx = blockIdx.x * blockDim.x + threadIdx.x;
    if (idx >= N_NODE * BCpad) return;
    const int n = idx / BCpad;
    const int col = idx % BCpad;
    float v = 0.f;
    if (col < BATCH * C) {
        const int b = col / C, c = col % C;
        if (c < Cin) {
            v = xsrc[((b * TT + t) * N_NODE + n) * Cin + c];
        } else {
            const int o = c - Cin;
            const float z = zr[(n * BATCH + b) * (2 * HID) + o];
            v = z * h[(n * BATCH + b) * HID + o];
        }
    }
    cnd[idx] = v;
}

// ---------------------------------------------------------------- diffusion GEMM
// Y(2048 x ncols) = Amat(2048x2048) @ X(2048 x ncols); ncols % 64 == 0.
// 4 waves, 64x64 tile, K-step 16 (4x WMMA f32 16x16x4 micro-steps).
// LDS tiles double-buffered via async global->LDS B128 copies (ASYNCcnt).
__global__ void __launch_bounds__(128)
gemm_AX_kernel(const float* __restrict__ Amat, const float* __restrict__ X,
               float* __restrict__ Y, int ncols) {
    // A-tile row stride 20 floats: chunks stay 16B-aligned for B128 LDS
    // writes, and 20*r mod 64 is distinct for r in [0,16) -> bank-clean
    // A-fragment reads.
    __shared__ float At[2][64][20];
    __shared__ float Xt[2][16][64];
    const int tid  = threadIdx.x;
    const int wave = tid >> 5;
    const int lane = tid & 31;
    const int row0 = blockIdx.x * 64;
    const int col0 = blockIdx.y * 64;

    // Async staging slots: 2 A-chunks + 2 X-chunks (16B each) per thread.
    unsigned aLds[2], xLds[2], aGof[2], xGof[2];
#pragma unroll
    for (int j = 0; j < 2; ++j) {
        const int ch = tid + 128 * j;                  // chunk id 0..255
        const int ar = ch >> 2, ak = (ch & 3) << 2;    // A: 64 rows x 4 chunks
        aLds[j] = (unsigned)(size_t)&At[0][ar][ak];
        aGof[j] = (unsigned)(((row0 + ar) * N_NODE + ak) * 4);
        const int xr = ch >> 4, xc = (ch & 15) << 2;   // X: 16 rows x 16 chunks
        xLds[j] = (unsigned)(size_t)&Xt[0][xr][xc];
        xGof[j] = (unsigned)((xr * ncols + col0 + xc) * 4);
    }
    const unsigned aBufStride = 64u * 20u * 4u;           // bytes
    const unsigned xBufStride = 16u * 64u * 4u;           // bytes
    const unsigned xKStride   = (unsigned)ncols * 16u * 4u; // bytes per K-step

    // preload buffer 0 (kb = 0)
#pragma unroll
    for (int j = 0; j < 2; ++j) {
        async_copy_b128(aLds[j], aGof[j], Amat);
        async_copy_b128(xLds[j], xGof[j], X);
    }

    v8f acc[4];
#pragma unroll
    for (int ct = 0; ct < 4; ++ct) acc[ct] = (v8f){0,0,0,0,0,0,0,0};

    const int mrow  = (wave << 4) + (lane & 15);
    const int khalf = (lane >> 4) << 1;          // 0 or 2
    const int ccol  = lane & 15;

    int buf = 0;
    for (int kb = 0; kb < N_NODE; kb += 16) {
        if (kb + 16 < N_NODE) {
            // prefetch next K-step into the other buffer
            const unsigned nb = (unsigned)(buf ^ 1);
            const unsigned kn = (unsigned)(kb + 16);
#pragma unroll
            for (int j = 0; j < 2; ++j) {
                async_copy_b128(aLds[j] + nb * aBufStride, aGof[j] + kn * 4u, Amat);
                async_copy_b128(xLds[j] + nb * xBufStride,
                                xGof[j] + (kn >> 4) * xKStride, X);
            }
            // async loads complete in order: <=4 outstanding => current buf done
            asm volatile("s_wait_asynccnt 0x4" ::: "memory");
        } else {
            asm volatile("s_wait_asynccnt 0x0" ::: "memory");
        }
        __syncthreads();
#pragma unroll
        for (int kk = 0; kk < 16; kk += 4) {
            const int k0 = kk + khalf;
            v2f a;
            a.x = At[buf][mrow][k0];
            a.y = At[buf][mrow][k0 + 1];
#pragma unroll
            for (int ct = 0; ct < 4; ++ct) {
                v2f bf;
                bf.x = Xt[buf][k0][(ct << 4) + ccol];
                bf.y = Xt[buf][k0 + 1][(ct << 4) + ccol];
                acc[ct] = __builtin_amdgcn_wmma_f32_16x16x4_f32(
                    false, a, false, bf, (short)0, acc[ct], false, false);
            }
        }
        __syncthreads();   // all waves done reading buf before it is re-filled
        buf ^= 1;
    }

    const int rowbase = row0 + (wave << 4) + ((lane >> 4) << 3);
    const int colbase = col0 + ccol;
#pragma unroll
    for (int ct = 0; ct < 4; ++ct)
#pragma unroll
        for (int v = 0; v < 8; ++v)
            Y[(size_t)(rowbase + v) * ncols + colbase + (ct << 4)] = acc[ct][v];
}

// ---------------------------------------------------------------- gconv apply
// One node per workgroup (4 waves). out[n][b][o] =
//   act( sum_e E[n,e] * (x_g[n] @ Wpool[e]) + E[n,:]@bpool )
// x_g rows = batch (16 = one WMMA M-tile), K = 2C, scaled A-frag trick.
__global__ void __launch_bounds__(128)
apply_gconv_kernel(const float* __restrict__ xs, const float* __restrict__ xsg,
                   const float* __restrict__ E,
                   const float* __restrict__ Wp, const float* __restrict__ bp,
                   float* __restrict__ out,
                   int C, int BCpad, int O, int act) {
    const int n    = blockIdx.x;
    const int tid  = threadIdx.x;
    const int wave = tid >> 5;
    const int lane = tid & 31;
    __shared__ float xg[16][257];   // pad 257: row-stride 256 would 16-way conflict
    __shared__ float erow[EMB];
    if (tid < EMB) erow[tid] = E[n * EMB + tid];

    const int KC = 2 * C;           // 132 (layer0) or 256 (layer1), both %4==0
    for (int i = tid; i < 16 * KC; i += 128) {
        const int b = i / KC, kc = i % KC;
        float v;
        if (kc < C) v = xs [n * BCpad + b * C + kc];
        else        v = xsg[n * BCpad + b * C + (kc - C)];
        xg[b][kc] = v;
    }
    __syncthreads();

    const int brow  = lane & 15;
    const int khalf = (lane >> 4) << 1;
    const int ocol  = lane & 15;
    const int nt    = O >> 4;

    for (int ot = wave; ot < nt; ot += 4) {      // wave-uniform: EXEC stays full
        const int o = (ot << 4) + ocol;
        v8f acc = (v8f){0,0,0,0,0,0,0,0};
        for (int e = 0; e < EMB; ++e) {
            const float s = erow[e];
            const float* We = Wp + (size_t)e * (CHEBK * C * O);
            for (int kc0 = 0; kc0 < KC; kc0 += 4) {
                const int k0 = kc0 + khalf;
                v2f a;
                a.x = xg[brow][k0] * s;          // fold E[n,e] into A operand
                a.y = xg[brow][k0 + 1] * s;
                v2f bf;
                {
                    const int kc = k0;
                    const int k = (kc >= C) ? 1 : 0;
                    bf.x = We[((size_t)(k * C + (kc - k * C))) * O + o];
                }
                {
                    const int kc = k0 + 1;
                    const int k = (kc >= C) ? 1 : 0;
                    bf.y = We[((size_t)(k * C + (kc - k * C))) * O + o];
                }
                acc = __builtin_amdgcn_wmma_f32_16x16x4_f32(
                    false, a, false, bf, (short)0, acc, false, false);
            }
        }
        float bias = 0.f;
#pragma unroll
        for (int e = 0; e < EMB; ++e) bias += erow[e] * bp[e * O + o];

#pragma unroll
        for (int v = 0; v < 8; ++v) {
            float xv = acc[v] + bias;
            xv = act ? tanhf(xv) : 1.0f / (1.0f + expf(-xv));
            const int b = v + ((lane >> 4) << 3);
            out[((size_t)n * BATCH + b) * O + o] = xv;
        }
    }
}

// ---------------------------------------------------------------- GRU update
__global__ void update_kernel(const float* __restrict__ zr,
                              const float* __restrict__ hc,
                              float* __restrict__ h,
                              float* __restrict__ outcur, int t) {
    const int idx = blockIdx.x * blockDim.x + threadIdx.x;
    if (idx >= N_NODE * BATCH * HID) return;
    const int o = idx % HID;
    const int b = (idx / HID) % BATCH;
    const int n = idx / (HID * BATCH);
    const float r  = zr[(n * BATCH + b) * (2 * HID) + HID + o];
    const float hn = r * h[idx] + (1.f - r) * hc[idx];
    h[idx] = hn;
    outcur[((size_t)(b * TT + t) * N_NODE + n) * HID + o] = hn;
}

__global__ void write_final_kernel(const float* __restrict__ h,
                                   float* __restrict__ finals, int layer) {
    const int idx = blockIdx.x * blockDim.x + threadIdx.x;
    if (idx >= N_NODE * BATCH * HID) return;
    const int o = idx % HID;
    const int b = (idx / HID) % BATCH;
    const int n = idx / (HID * BATCH);
    finals[((size_t)(layer * BATCH + b) * N_NODE + n) * HID + o] = h[idx];
}

// ---------------------------------------------------------------- launch
extern "C" void kernel_launch(void* const* d_in, const int* in_sizes, int n_in,
                              void* d_out, int out_size, void* d_ws, size_t ws_size,
                              hipStream_t stream) {
    (void)in_sizes; (void)n_in; (void)out_size; (void)ws_size;
    const float* x   = (const float*)d_in[0];
    const float* ini = (const float*)d_in[1];
    const float* E   = (const float*)d_in[2];
    const float* Wg[2] = {(const float*)d_in[3], (const float*)d_in[7]};
    const float* bg[2] = {(const float*)d_in[4], (const float*)d_in[8]};
    const float* Wu[2] = {(const float*)d_in[5], (const float*)d_in[9]};
    const float* bu[2] = {(const float*)d_in[6], (const float*)d_in[10]};

    float* outcur = (float*)d_out;                                    // (B,T,N,64)
    float* finals = outcur + (size_t)BATCH * TT * N_NODE * HID;       // (2,B,N,64)

    float* ws = (float*)d_ws;
    size_t off = 0;
    float* A    = ws + off; off += (size_t)N_NODE * N_NODE;           // 16.8 MB
    float* xsb  = ws + off; off += (size_t)N_NODE * 2048;
    float* xsg  = ws + off; off += (size_t)N_NODE * 2048;
    float* cnd  = ws + off; off += (size_t)N_NODE * 2048;
    float* cndg = ws + off; off += (size_t)N_NODE * 2048;
    float* zr   = ws + off; off += (size_t)N_NODE * BATCH * 2 * HID;
    float* hc   = ws + off; off += (size_t)N_NODE * BATCH * HID;
    float* hbuf = ws + off; off += (size_t)N_NODE * BATCH * HID;      // ~118 MB total

    supports_kernel<<<N_NODE, 256, 0, stream>>>(E, A);

    const int NBH  = N_NODE * BATCH * HID;
    const int ebg  = (NBH + 255) / 256;

    for (int layer = 0; layer < 2; ++layer) {
        const int Cin   = (layer == 0) ? 2 : HID;
        const int C     = Cin + HID;                    // 66 / 128
        const int BCpad = (layer == 0) ? 1088 : 2048;   // round B*C up to 64
        const float* xsrc = (layer == 0) ? x : outcur;  // layer1 reads layer0 seq in-place

        init_h_kernel<<<ebg, 256, 0, stream>>>(ini, hbuf, layer);

        const int bx_total = N_NODE * BCpad;
        const int bx_blocks = (bx_total + 255) / 256;
        dim3 ggrid(N_NODE / 64, BCpad / 64);

        for (int t = 0; t < TT; ++t) {
            build_xs_kernel<<<bx_blocks, 256, 0, stream>>>(xsrc, Cin, t, hbuf, xsb, C, BCpad);
            gemm_AX_kernel<<<ggrid, 128, 0, stream>>>(A, xsb, xsg, BCpad);
            apply_gconv_kernel<<<N_NODE, 128, 0, stream>>>(xsb, xsg, E, Wg[layer], bg[layer],
                                                           zr, C, BCpad, 2 * HID, /*act=*/0);
            build_cnd_kernel<<<bx_blocks, 256, 0, stream>>>(xsrc, Cin, t, hbuf, zr, cnd, C, BCpad);
            gemm_AX_kernel<<<ggrid, 128, 0, stream>>>(A, cnd, cndg, BCpad);
            apply_gconv_kernel<<<N_NODE, 128, 0, stream>>>(cnd, cndg, E, Wu[layer], bu[layer],
                                                           hc, C, BCpad, HID, /*act=*/1);
            update_kernel<<<ebg, 256, 0, stream>>>(zr, hc, hbuf, outcur, t);
        }
        write_final_kernel<<<ebg, 256, 0, stream>>>(hbuf, finals, layer);
    }
}